// SAGEConv_62878321213496
// MI455X (gfx1250) — compile-verified
//
#include <hip/hip_runtime.h>

typedef float v2f __attribute__((ext_vector_type(2)));
typedef float v8f __attribute__((ext_vector_type(8)));

#define DF 64  // DIN == DOUT == 64

// ---------------------------------------------------------------------------
// WMMA helper: D = A(16x4,f32) x B(4x16,f32) + C(16x16,f32)
// ---------------------------------------------------------------------------
__device__ __forceinline__ v8f wmma4(v2f a, v2f b, v8f c) {
  // 8 args: (neg_a, A, neg_b, B, c_mod, C, reuse_a, reuse_b)
  return __builtin_amdgcn_wmma_f32_16x16x4_f32(
      false, a, false, b, (short)0, c, false, false);
}

// Compute one 16x16 f32 tile of feat_tile @ W^T.
// A[16]  : per-lane preloaded A fragments for K chunks 0..60 step 4
// Wrow   : W + (n0+col)*DF + 2*half  (per-lane base into weight matrix)
__device__ __forceinline__ v8f gemm_tile(const v2f* A, const float* __restrict__ Wrow) {
  v8f c = {};
#pragma unroll
  for (int kc = 0; kc < 16; ++kc) {
    v2f b = *(const v2f*)(Wrow + kc * 4);
    c = wmma4(A[kc], b, c);
  }
  return c;
}

// ---------------------------------------------------------------------------
// Kernel 1: zero the accumulators (agg_low | agg_high | deg are contiguous)
// ---------------------------------------------------------------------------
__global__ __launch_bounds__(256) void zero_kernel(float* __restrict__ p, size_t n) {
  size_t i = (size_t)blockIdx.x * blockDim.x + threadIdx.x;
  size_t stride = (size_t)gridDim.x * blockDim.x;
  for (; i < n; i += stride) p[i] = 0.0f;
}

// ---------------------------------------------------------------------------
// Kernel 2: msg_{low,high}[i] = relu(feat[i] @ W^T + b) * norm[i]
// One wave (32 threads) per 16-node tile. WMMA f32 16x16x4.
// ---------------------------------------------------------------------------
__global__ __launch_bounds__(32) void msg_kernel(
    const float* __restrict__ feat, const float* __restrict__ norm,
    const float* __restrict__ Wn, const float* __restrict__ bn,
    const float* __restrict__ Wh, const float* __restrict__ bh,
    float* __restrict__ msg_low, float* __restrict__ msg_high) {
  const int lane = threadIdx.x;
  const int col  = lane & 15;
  const int half = lane >> 4;
  const int m0   = blockIdx.x << 4;

  // Preload A fragments: lane holds feat[m0+col][k0 + 2*half + {0,1}]
  const float* arow = feat + (size_t)(m0 + col) * DF + 2 * half;
  v2f A[16];
#pragma unroll
  for (int kc = 0; kc < 16; ++kc) A[kc] = *(const v2f*)(arow + kc * 4);

  float nrm[8];
#pragma unroll
  for (int j = 0; j < 8; ++j) nrm[j] = norm[m0 + j + 8 * half];

#pragma unroll
  for (int nt = 0; nt < 4; ++nt) {
    const int n0 = nt << 4;
    const int n  = n0 + col;
    {  // low branch (W_neigh)
      v8f c = gemm_tile(A, Wn + (size_t)n * DF + 2 * half);
      const float bias = bn[n];
#pragma unroll
      for (int j = 0; j < 8; ++j) {
        const int row = m0 + j + 8 * half;
        float v = c[j] + bias;
        v = v > 0.0f ? v : 0.0f;
        msg_low[(size_t)row * DF + n] = v * nrm[j];
      }
    }
    {  // high branch (W_neigh_high)
      v8f c = gemm_tile(A, Wh + (size_t)n * DF + 2 * half);
      const float bias = bh[n];
#pragma unroll
      for (int j = 0; j < 8; ++j) {
        const int row = m0 + j + 8 * half;
        float v = c[j] + bias;
        v = v > 0.0f ? v : 0.0f;
        msg_high[(size_t)row * DF + n] = v * nrm[j];
      }
    }
  }
}

// ---------------------------------------------------------------------------
// Kernel 3: edge scatter. 64 threads per edge, one channel each.
// agg_*[dst] += msg_*[src]; deg[dst] += 1
// ---------------------------------------------------------------------------
__global__ __launch_bounds__(256) void scatter_kernel(
    const int* __restrict__ src, const int* __restrict__ dst,
    const float* __restrict__ msg_low, const float* __restrict__ msg_high,
    float* __restrict__ agg_low, float* __restrict__ agg_high,
    float* __restrict__ deg, int E) {
  const long long t = (long long)blockIdx.x * 256 + threadIdx.x;
  const int e  = (int)(t >> 6);
  const int ch = (int)(t & 63);
  if (e >= E) return;
  const int s = src[e];
  const int d = dst[e];
  unsafeAtomicAdd(&agg_low[(size_t)d * DF + ch], msg_low[(size_t)s * DF + ch]);
  unsafeAtomicAdd(&agg_high[(size_t)d * DF + ch], msg_high[(size_t)s * DF + ch]);
  if (ch == 0) unsafeAtomicAdd(&deg[d], 1.0f);
}

// ---------------------------------------------------------------------------
// Kernel 4: self/identity WMMA linears + combine + attention + output.
// One wave per 16-node tile; branch tiles staged in padded LDS.
// ---------------------------------------------------------------------------
__global__ __launch_bounds__(32) void out_kernel(
    const float* __restrict__ feat,
    const float* __restrict__ Ws,  const float* __restrict__ bs,
    const float* __restrict__ Wsh, const float* __restrict__ bsh,
    const float* __restrict__ Wid, const float* __restrict__ bid,
    const float* __restrict__ agg_low, const float* __restrict__ agg_high,
    const float* __restrict__ deg,
    const float* __restrict__ w_low, const float* __restrict__ w_high,
    const float* __restrict__ w_mlp, const float* __restrict__ W_att,
    float* __restrict__ out) {
  __shared__ float lowS[16 * 68];   // pad to 68 floats/row: no bank conflicts
  __shared__ float highS[16 * 68];
  __shared__ float idnS[16 * 68];
  __shared__ float att[16 * 4];

  const int lane = threadIdx.x;
  const int col  = lane & 15;
  const int half = lane >> 4;
  const int m0   = blockIdx.x << 4;

  const float* arow = feat + (size_t)(m0 + col) * DF + 2 * half;
  v2f A[16];
#pragma unroll
  for (int kc = 0; kc < 16; ++kc) A[kc] = *(const v2f*)(arow + kc * 4);

  float invd[8];
#pragma unroll
  for (int j = 0; j < 8; ++j) {
    const float dg = deg[m0 + j + 8 * half];
    invd[j] = 1.0f / (dg > 1.0f ? dg : 1.0f);
  }

#pragma unroll
  for (int nt = 0; nt < 4; ++nt) {
    const int n0 = nt << 4;
    const int n  = n0 + col;
    {  // low = relu(feat@W_self^T+b) + agg_low/deg
      v8f c = gemm_tile(A, Ws + (size_t)n * DF + 2 * half);
      const float bias = bs[n];
#pragma unroll
      for (int j = 0; j < 8; ++j) {
        const int row = m0 + j + 8 * half;
        float v = c[j] + bias;
        v = v > 0.0f ? v : 0.0f;
        v += agg_low[(size_t)row * DF + n] * invd[j];
        lowS[(j + 8 * half) * 68 + n] = v;
      }
    }
    {  // high = relu(feat@W_self_high^T+b) - agg_high/deg
      v8f c = gemm_tile(A, Wsh + (size_t)n * DF + 2 * half);
      const float bias = bsh[n];
#pragma unroll
      for (int j = 0; j < 8; ++j) {
        const int row = m0 + j + 8 * half;
        float v = c[j] + bias;
        v = v > 0.0f ? v : 0.0f;
        v -= agg_high[(size_t)row * DF + n] * invd[j];
        highS[(j + 8 * half) * 68 + n] = v;
      }
    }
    {  // identity = relu(feat@W_id^T+b)
      v8f c = gemm_tile(A, Wid + (size_t)n * DF + 2 * half);
      const float bias = bid[n];
#pragma unroll
      for (int j = 0; j < 8; ++j) {
        float v = c[j] + bias;
        v = v > 0.0f ? v : 0.0f;
        idnS[(j + 8 * half) * 68 + n] = v;
      }
    }
  }
  __syncthreads();

  if (lane < 16) {  // one row per lane: attention scalars
    float d0 = 0.0f, d1 = 0.0f, d2 = 0.0f;
    for (int c2 = 0; c2 < DF; ++c2) {
      d0 += lowS[lane * 68 + c2]  * w_low[c2];
      d1 += highS[lane * 68 + c2] * w_high[c2];
      d2 += idnS[lane * 68 + c2]  * w_mlp[c2];
    }
    const float s0 = 1.0f / (1.0f + __expf(-d0));
    const float s1 = 1.0f / (1.0f + __expf(-d1));
    const float s2 = 1.0f / (1.0f + __expf(-d2));
    const float t0 = (W_att[0] * s0 + W_att[1] * s1 + W_att[2] * s2) * (1.0f / 3.0f);
    const float t1 = (W_att[3] * s0 + W_att[4] * s1 + W_att[5] * s2) * (1.0f / 3.0f);
    const float t2 = (W_att[6] * s0 + W_att[7] * s1 + W_att[8] * s2) * (1.0f / 3.0f);
    const float mx = fmaxf(t0, fmaxf(t1, t2));
    const float e0 = __expf(t0 - mx), e1 = __expf(t1 - mx), e2 = __expf(t2 - mx);
    const float inv3 = 3.0f / (e0 + e1 + e2);  // fold the 3.0x scale into att
    att[lane * 4 + 0] = e0 * inv3;
    att[lane * 4 + 1] = e1 * inv3;
    att[lane * 4 + 2] = e2 * inv3;
  }
  __syncthreads();

  // Coalesced output: 16 rows x 64 cols = 1024 floats, 32 per lane.
#pragma unroll
  for (int i = 0; i < 32; ++i) {
    const int idx = lane + 32 * i;
    const int r = idx >> 6, cc = idx & 63;
    const float v = att[r * 4 + 0] * lowS[r * 68 + cc] +
                    att[r * 4 + 1] * highS[r * 68 + cc] +
                    att[r * 4 + 2] * idnS[r * 68 + cc];
    out[(size_t)m0 * DF + idx] = v;
  }
}

// ---------------------------------------------------------------------------
// Host launcher
// ---------------------------------------------------------------------------
extern "C" void kernel_launch(void* const* d_in, const int* in_sizes, int n_in,
                              void* d_out, int out_size, void* d_ws, size_t ws_size,
                              hipStream_t stream) {
  const float* feat         = (const float*)d_in[0];
  const float* norm         = (const float*)d_in[1];
  const int*   src          = (const int*)d_in[2];
  const int*   dst          = (const int*)d_in[3];
  const float* W_self       = (const float*)d_in[4];
  const float* b_self       = (const float*)d_in[5];
  const float* W_self_high  = (const float*)d_in[6];
  const float* b_self_high  = (const float*)d_in[7];
  const float* W_neigh      = (const float*)d_in[8];
  const float* b_neigh      = (const float*)d_in[9];
  const float* W_neigh_high = (const float*)d_in[10];
  const float* b_neigh_high = (const float*)d_in[11];
  const float* W_id         = (const float*)d_in[12];
  const float* b_id         = (const float*)d_in[13];
  const float* w_att_low    = (const float*)d_in[14];
  const float* w_att_high   = (const float*)d_in[15];
  const float* w_att_mlp    = (const float*)d_in[16];
  const float* W_att        = (const float*)d_in[17];
  float* out = (float*)d_out;

  const int N = in_sizes[1];  // norm has N elements
  const int E = in_sizes[2];  // src has E elements

  // Workspace layout: msg_low | msg_high | agg_low | agg_high | deg
  float* msg_low  = (float*)d_ws;
  float* msg_high = msg_low  + (size_t)N * DF;
  float* agg_low  = msg_high + (size_t)N * DF;
  float* agg_high = agg_low  + (size_t)N * DF;
  float* degp     = agg_high + (size_t)N * DF;

  const size_t zero_n = (size_t)N * DF * 2 + (size_t)N;  // agg_low..deg contiguous
  zero_kernel<<<2048, 256, 0, stream>>>(agg_low, zero_n);

  msg_kernel<<<N / 16, 32, 0, stream>>>(feat, norm, W_neigh, b_neigh,
                                        W_neigh_high, b_neigh_high,
                                        msg_low, msg_high);

  const long long work = (long long)E * DF;
  scatter_kernel<<<(unsigned)((work + 255) / 256), 256, 0, stream>>>(
      src, dst, msg_low, msg_high, agg_low, agg_high, degp, E);

  out_kernel<<<N / 16, 32, 0, stream>>>(feat, W_self, b_self,
                                        W_self_high, b_self_high,
                                        W_id, b_id,
                                        agg_low, agg_high, degp,
                                        w_att_low, w_att_high, w_att_mlp, W_att,
                                        out);
}